// VNResnetPointnet_31035433681598
// MI455X (gfx1250) — compile-verified
//
#include <hip/hip_runtime.h>
#include <hip/hip_bf16.h>

typedef float v2f __attribute__((ext_vector_type(2)));
typedef float v8f __attribute__((ext_vector_type(8)));

#define BB   8
#define NN   2048
#define KNN  20
#define CIN  256
#define H3   128
#define MTOT (BB * 3 * NN)   // 49152
#define EPSF 1e-6f

// ---------------------------------------------------------------------------
// CDNA5 async copy helpers (CDNA5 ISA: GLOBAL_LOAD_ASYNC_TO_LDS_*).
// INST_OFFSET is added to BOTH the LDS and the global address, so one base
// address pair serves four contiguous b128 transfers.
// ---------------------------------------------------------------------------
__device__ __forceinline__ void async_ld_b128x4(unsigned lds_addr, const float* g) {
    unsigned long long ga = (unsigned long long)g;
    asm volatile("global_load_async_to_lds_b128 %0, %1, off"
                 :: "v"(lds_addr), "v"(ga) : "memory");
    asm volatile("global_load_async_to_lds_b128 %0, %1, off offset:16"
                 :: "v"(lds_addr), "v"(ga) : "memory");
    asm volatile("global_load_async_to_lds_b128 %0, %1, off offset:32"
                 :: "v"(lds_addr), "v"(ga) : "memory");
    asm volatile("global_load_async_to_lds_b128 %0, %1, off offset:48"
                 :: "v"(lds_addr), "v"(ga) : "memory");
}

__device__ __forceinline__ void wait_async0() {
    asm volatile("s_wait_asynccnt 0" ::: "memory");
}

// ---------------------------------------------------------------------------
// Stage 1: kNN + mean-neighbor -> per-point 3x3 feature matrix G
// (mean over k commutes with all linears, so only mean neighbor pos needed)
// ---------------------------------------------------------------------------
__global__ __launch_bounds__(256) void knn_feat_kernel(const float* __restrict__ p,
                                                       float* __restrict__ G) {
    __shared__ float sx[NN], sy[NN], sz[NN];
    const int b    = blockIdx.x;
    const int tile = blockIdx.y;
    const int lane = threadIdx.x & 31;
    const int wave = threadIdx.x >> 5;

    const float* pb = p + (size_t)b * NN * 3;
    for (int i = threadIdx.x; i < NN; i += 256) {
        sx[i] = pb[i * 3 + 0];
        sy[i] = pb[i * 3 + 1];
        sz[i] = pb[i * 3 + 2];
    }
    __syncthreads();

    const int n = tile * 8 + wave;
    const float cx = sx[n], cy = sy[n], cz = sz[n];

    float dist[64];
#pragma unroll
    for (int t = 0; t < 64; ++t) {
        const int m = t * 32 + lane;
        const float mx = sx[m], my = sy[m], mz = sz[m];
        dist[t] = 2.0f * (cx * mx + cy * my + cz * mz) - (mx * mx + my * my + mz * mz);
    }

    unsigned long long consumed = 0ull;
    float axv = 0.f, ayv = 0.f, azv = 0.f;

    for (int s = 0; s < KNN; ++s) {
        float best = -3.4e38f;
        int   bidx = 0x7fffffff;
#pragma unroll
        for (int t = 0; t < 64; ++t) {
            const bool dead = (consumed >> t) & 1ull;
            const float v = dead ? -3.0e38f : dist[t];
            const int idx = t * 32 + lane;
            if (v > best || (v == best && idx < bidx)) { best = v; bidx = idx; }
        }
#pragma unroll
        for (int off = 16; off > 0; off >>= 1) {
            const float ov = __shfl_xor(best, off, 32);
            const int   oi = __shfl_xor(bidx, off, 32);
            if (ov > best || (ov == best && oi < bidx)) { best = ov; bidx = oi; }
        }
        if (lane == (bidx & 31)) consumed |= (1ull << (bidx >> 5));
        axv += sx[bidx]; ayv += sy[bidx]; azv += sz[bidx];
    }

    if (lane == 0) {
        const float inv = 1.0f / (float)KNN;
        const float mx = axv * inv, my = ayv * inv, mz = azv * inv;
        float* g = G + ((size_t)b * NN + n) * 9;
        g[0] = mx - cx; g[1] = my - cy; g[2] = mz - cz;
        g[3] = cx;      g[4] = cy;      g[5] = cz;
        g[6] = my * cz - mz * cy;
        g[7] = mz * cx - mx * cz;
        g[8] = mx * cy - my * cx;
    }
}

// ---------------------------------------------------------------------------
// Stage 2: x[c][b][d][n] = sum_s Wpos[c][s] * G[b][n][s][d]
// ---------------------------------------------------------------------------
__global__ void pos_embed_kernel(const float* __restrict__ Wpos,
                                 const float* __restrict__ G,
                                 float* __restrict__ X) {
    const int gid = blockIdx.x * blockDim.x + threadIdx.x;
    if (gid >= CIN * BB * NN) return;
    const int n = gid % NN;
    const int t = gid / NN;
    const int b = t % BB;
    const int o = t / BB;
    const float* g = G + ((size_t)b * NN + n) * 9;
    const float w0 = Wpos[o * 3 + 0], w1 = Wpos[o * 3 + 1], w2 = Wpos[o * 3 + 2];
    const size_t base = ((size_t)(o * BB + b) * 3) * NN + n;
    X[base]          = w0 * g[0] + w1 * g[3] + w2 * g[6];
    X[base + NN]     = w0 * g[1] + w1 * g[4] + w2 * g[7];
    X[base + 2 * NN] = w0 * g[2] + w1 * g[5] + w2 * g[8];
}

// ---------------------------------------------------------------------------
// WMMA f32 GEMM:  Y(O x MTOT) (+)= W(O x C) * X(C x MTOT)
// 8 waves as 2(O) x 4(M); wave tile 16x64; K chunked by 16.
// X tiles streamed global->LDS with GLOBAL_LOAD_ASYNC_TO_LDS_B128 (ASYNCcnt),
// double-buffered; A panel prefetched to registers per chunk.
// V_WMMA_F32_16X16X4_F32 lane layout (ISA 7.12.2):
//   A 16x4 : lane l holds row l%16, K = k + 2*(l>>4) + {0,1}
//   B 4x16 : lane l holds col l%16, K = k + 2*(l>>4) + {0,1}
//   C/D    : lane l, vgpr v -> row v + 8*(l>>4), col l%16
// ---------------------------------------------------------------------------
#define XS_STRIDE 264   // 16 rows x 264 floats; pad keeps k / k+2 rows on
                        // disjoint LDS bank groups for the two half-waves

template <int O, int C, bool ACC>
__global__ __launch_bounds__(256) void gemm_wmma_kernel(const float* __restrict__ W,
                                                        const float* __restrict__ X,
                                                        float* __restrict__ Y) {
    constexpr int M   = MTOT;
    constexpr int NCH = C / 16;
    __shared__ __align__(16) float xs[2][16 * XS_STRIDE];

    const int tid  = threadIdx.x;
    const int lane = tid & 31;
    const int wave = tid >> 5;
    const int l16  = lane & 15;
    const int hi   = lane >> 4;
    const int otile = blockIdx.x * 32 + (wave >> 2) * 16;
    const int mblk  = blockIdx.y * 256;
    const int mloc  = (wave & 3) * 64;
    const int arow  = otile + l16;

    // cooperative async-loader mapping: thread -> (row, 16-col group) of tile
    const int lrow = tid >> 4;          // 0..15
    const int lcol = (tid & 15) * 16;   // 0..240
    const unsigned lds0 = (unsigned)(unsigned long long)(const void*)&xs[0][0];

    auto issue = [&](int chunk, int buf) {
        const float* gsrc = X + (size_t)(chunk * 16 + lrow) * M + mblk + lcol;
        const unsigned ldst =
            lds0 + (unsigned)((buf * 16 * XS_STRIDE + lrow * XS_STRIDE + lcol) * 4);
        async_ld_b128x4(ldst, gsrc);
    };

    v8f acc[4];
    if (ACC) {
#pragma unroll
        for (int j = 0; j < 4; ++j) {
            const int col = mblk + mloc + j * 16 + l16;
#pragma unroll
            for (int v = 0; v < 8; ++v)
                acc[j][v] = Y[(size_t)(otile + v + 8 * hi) * M + col];
        }
    } else {
#pragma unroll
        for (int j = 0; j < 4; ++j) acc[j] = (v8f)0.0f;
    }

    issue(0, 0);
    for (int ch = 0; ch < NCH; ++ch) {
        wait_async0();          // this wave's tile landed in LDS
        __syncthreads();        // every wave's tile landed; prev reads done

        if (ch + 1 < NCH) issue(ch + 1, (ch + 1) & 1);

        // A panel prefetch: 4 k-steps of this chunk (clause-able b64 loads)
        const int kc = ch * 16;
        v2f a[4];
#pragma unroll
        for (int s = 0; s < 4; ++s) {
            const float* wp = W + (size_t)arow * C + kc + 4 * s + 2 * hi;
            a[s].x = wp[0];
            a[s].y = wp[1];
        }

        const float* xb = &xs[ch & 1][0];
#pragma unroll
        for (int s = 0; s < 4; ++s) {
            const int kl = 4 * s + 2 * hi;
            const float* xr = xb + kl * XS_STRIDE;
#pragma unroll
            for (int j = 0; j < 4; ++j) {
                const int col = mloc + j * 16 + l16;
                v2f bv;
                bv.x = xr[col];
                bv.y = xr[col + XS_STRIDE];
                acc[j] = __builtin_amdgcn_wmma_f32_16x16x4_f32(
                    false, a[s], false, bv, (short)0, acc[j], false, false);
            }
        }
    }

#pragma unroll
    for (int j = 0; j < 4; ++j) {
        const int col = mblk + mloc + j * 16 + l16;
#pragma unroll
        for (int v = 0; v < 8; ++v)
            Y[(size_t)(otile + v + 8 * hi) * M + col] = acc[j][v];
    }
}

// ---------------------------------------------------------------------------
// VN-ReLU: couples the 3 vector components per channel; in-place into D.
// ---------------------------------------------------------------------------
__global__ void vn_relu_kernel(const float* __restrict__ X,
                               float* __restrict__ D, int Cch) {
    const int gid = blockIdx.x * blockDim.x + threadIdx.x;
    if (gid >= Cch * BB * NN) return;
    const int n = gid % NN;
    const int t = gid / NN;
    const int b = t % BB;
    const int c = t / BB;
    const size_t base = ((size_t)(c * BB + b) * 3) * NN + n;
    float x0 = X[base], x1 = X[base + NN], x2 = X[base + 2 * NN];
    const float d0 = D[base], d1 = D[base + NN], d2 = D[base + 2 * NN];
    const float dot = x0 * d0 + x1 * d1 + x2 * d2;
    if (dot < 0.0f) {
        const float dsq = d0 * d0 + d1 * d1 + d2 * d2;
        const float s = dot / (dsq + EPSF);
        x0 -= s * d0; x1 -= s * d1; x2 -= s * d2;
    }
    D[base] = x0; D[base + NN] = x1; D[base + 2 * NN] = x2;
}

__global__ __launch_bounds__(256) void pool_mean_kernel(const float* __restrict__ X,
                                                        float* __restrict__ P) {
    const int row  = blockIdx.x * 8 + (threadIdx.x >> 5);
    const int lane = threadIdx.x & 31;
    const float* xr = X + (size_t)row * NN;
    float s = 0.f;
    for (int i = lane; i < NN; i += 32) s += xr[i];
#pragma unroll
    for (int off = 16; off > 0; off >>= 1) s += __shfl_xor(s, off, 32);
    if (lane == 0) P[row] = s * (1.0f / (float)NN);
}

__global__ void concat_kernel(const float* __restrict__ OUT,
                              const float* __restrict__ P,
                              float* __restrict__ XN) {
    const int gid = blockIdx.x * blockDim.x + threadIdx.x;
    if (gid >= CIN * MTOT) return;
    const int m  = gid % MTOT;
    const int cc = gid / MTOT;
    if (cc < H3) XN[gid] = OUT[(size_t)cc * MTOT + m];
    else         XN[gid] = P[(cc - H3) * (BB * 3) + (m / NN)];
}

__global__ __launch_bounds__(128) void head_kernel(const float* __restrict__ Pn,
                                                   const float* __restrict__ Wd,
                                                   const float* __restrict__ Wc,
                                                   float* __restrict__ out) {
    const int b = blockIdx.x;
    const int o = threadIdx.x;
    __shared__ float ys[H3][3];
    float a0 = 0.f, a1 = 0.f, a2 = 0.f;
    for (int c = 0; c < H3; ++c) {
        const float w = Wd[o * H3 + c];
        const float* q = Pn + c * (BB * 3) + b * 3;
        a0 += w * q[0]; a1 += w * q[1]; a2 += w * q[2];
    }
    const float* xq = Pn + o * (BB * 3) + b * 3;
    float x0 = xq[0], x1 = xq[1], x2 = xq[2];
    const float dot = x0 * a0 + x1 * a1 + x2 * a2;
    if (dot < 0.0f) {
        const float dsq = a0 * a0 + a1 * a1 + a2 * a2;
        const float s = dot / (dsq + EPSF);
        x0 -= s * a0; x1 -= s * a1; x2 -= s * a2;
    }
    ys[o][0] = x0; ys[o][1] = x1; ys[o][2] = x2;
    __syncthreads();
    float c0 = 0.f, c1 = 0.f, c2 = 0.f;
    for (int c = 0; c < H3; ++c) {
        const float w = Wc[o * H3 + c];
        c0 += w * ys[c][0]; c1 += w * ys[c][1]; c2 += w * ys[c][2];
    }
    out[(b * H3 + o) * 3 + 0] = c0;
    out[(b * H3 + o) * 3 + 1] = c1;
    out[(b * H3 + o) * 3 + 2] = c2;
}

// ---------------------------------------------------------------------------
extern "C" void kernel_launch(void* const* d_in, const int* in_sizes, int n_in,
                              void* d_out, int out_size, void* d_ws, size_t ws_size,
                              hipStream_t stream) {
    (void)in_sizes; (void)n_in; (void)out_size; (void)ws_size;
    const float* p     = (const float*)d_in[0];
    const float* W_pos = (const float*)d_in[1];
    const float* W0s   = (const float*)d_in[2];
    const float* Wd0s  = (const float*)d_in[3];
    const float* W1s   = (const float*)d_in[4];
    const float* Wd1s  = (const float*)d_in[5];
    const float* Wss   = (const float*)d_in[6];
    const float* Wdact = (const float*)d_in[7];
    const float* W_c   = (const float*)d_in[8];
    float* out = (float*)d_out;

    float* A  = (float*)d_ws;                    // x       : CIN x MTOT
    float* Bt = A  + (size_t)CIN * MTOT;         // d0/y0   : CIN x MTOT
    float* Cn = Bt + (size_t)CIN * MTOT;         // netlin  : H3  x MTOT
    float* Dd = Cn + (size_t)H3  * MTOT;         // d1/y1   : H3  x MTOT
    float* E  = Dd + (size_t)H3  * MTOT;         // skip/out: H3  x MTOT
    float* G  = E  + (size_t)H3  * MTOT;         // 3x3 feats: B*N*9
    float* P  = G  + (size_t)BB * NN * 9;        // pooled  : H3*B*3

    const dim3 gridM(0, MTOT / 256);

    auto gemm = [&](const float* W, const float* X, float* Y, int O, int C, int acc) {
        dim3 grid(O / 32, MTOT / 256);
        if (O == 256 && C == 256)
            gemm_wmma_kernel<256, 256, false><<<grid, 256, 0, stream>>>(W, X, Y);
        else if (O == 128 && C == 256)
            gemm_wmma_kernel<128, 256, false><<<grid, 256, 0, stream>>>(W, X, Y);
        else if (O == 128 && C == 128 && !acc)
            gemm_wmma_kernel<128, 128, false><<<grid, 256, 0, stream>>>(W, X, Y);
        else
            gemm_wmma_kernel<128, 128, true><<<grid, 256, 0, stream>>>(W, X, Y);
    };

    knn_feat_kernel<<<dim3(BB, NN / 8), 256, 0, stream>>>(p, G);
    pos_embed_kernel<<<(CIN * BB * NN + 255) / 256, 256, 0, stream>>>(W_pos, G, A);

    for (int i = 0; i < 5; ++i) {
        const float* W0  = W0s  + (size_t)i * H3  * CIN;
        const float* Wd0 = Wd0s + (size_t)i * CIN * CIN;
        const float* W1  = W1s  + (size_t)i * H3  * H3;
        const float* Wd1 = Wd1s + (size_t)i * H3  * H3;
        const float* Ws  = Wss  + (size_t)i * H3  * CIN;

        gemm(Ws,  A,  E,  H3,  CIN, 0);                                   // skip = Ws @ x
        gemm(Wd0, A,  Bt, CIN, CIN, 0);                                   // d0   = Wd0 @ x
        vn_relu_kernel<<<(CIN * BB * NN + 255) / 256, 256, 0, stream>>>(A, Bt, CIN);
        gemm(W0,  Bt, Cn, H3,  CIN, 0);                                   // net  = W0 @ y0
        gemm(Wd1, Cn, Dd, H3,  H3,  0);                                   // d1   = Wd1 @ net
        vn_relu_kernel<<<(H3 * BB * NN + 255) / 256, 256, 0, stream>>>(Cn, Dd, H3);
        gemm(W1,  Dd, E,  H3,  H3,  1);                                   // out  = skip + W1 @ y1
        if (i < 4) {
            pool_mean_kernel<<<(H3 * BB * 3) / 8, 256, 0, stream>>>(E, P);
            concat_kernel<<<(CIN * MTOT + 255) / 256, 256, 0, stream>>>(E, P, A);
        }
    }

    pool_mean_kernel<<<(H3 * BB * 3) / 8, 256, 0, stream>>>(E, P);
    head_kernel<<<BB, H3, 0, stream>>>(P, Wdact, W_c, out);
}